// COMALoss_72885595013509
// MI455X (gfx1250) — compile-verified
//
#include <hip/hip_runtime.h>

#define TPB 256
#define NBLK 2048
// Grid stride is a compile-time power of two (2048*256 = 2^19) so the
// stage-count division folds to a shift instead of a runtime s_div sequence.
constexpr long long STRIDE = (long long)NBLK * (long long)TPB;

// ---------------------------------------------------------------------------
// LDS byte-offset helper: AS3 pointers on AMDGPU are 32-bit offsets from the
// wave's LDS base, which is exactly what global_load_async_to_lds_* wants in
// its LDS-address VGPR. Cast through u64 so the host compilation pass (where
// pointers are 64-bit) doesn't reject the truncation.
// ---------------------------------------------------------------------------
typedef __attribute__((address_space(3))) const void* lds_cptr_t;

__device__ __forceinline__ unsigned lds_byte_off(const void* p) {
#if defined(__HIP_DEVICE_COMPILE__)
  return (unsigned)(unsigned long long)(lds_cptr_t)p;
#else
  return 0u;
#endif
}

// CDNA5 async global -> LDS copy (128-bit per lane), tracked by ASYNCcnt.
__device__ __forceinline__ void async_copy_b128(const float4* g, unsigned lds_off) {
  asm volatile("global_load_async_to_lds_b128 %0, %1, off"
               :
               : "v"(lds_off), "v"(g)
               : "memory");
}

// ---------------------------------------------------------------------------
// Per-element contribution (pre-scale):
//   t = log_prob * p_flip * (sq_new - sq_old)*HW
// with h in {0,1}:  sq_new - sq_old = (1-2h)(1-2c)
//   p_flip  = h ? (1-p) : p
//   logarg  = h ? p : (1-p+eps)
// Final loss = -(1/N) * sum(t),  N = B*H*W  (per-batch means cancel).
// ---------------------------------------------------------------------------
__device__ __forceinline__ float coma_elem(float p, float c, float h) {
  const float s   = (1.0f - 2.0f * h) * (1.0f - 2.0f * c);
  const bool  hb  = (h != 0.0f);
  const float pf  = hb ? (1.0f - p) : p;
  const float arg = hb ? p : (1.0f - p + 1e-8f);
  return __logf(arg) * pf * s;
}

// ---------------------------------------------------------------------------
// Main streaming kernel: double-buffered async global->LDS staging.
// Each lane stages its own 16B slice of each of the 3 arrays, so no
// workgroup barriers are required; ASYNCcnt waits pipeline the stages.
// Must be launched with <<<NBLK, TPB>>> (STRIDE is baked in).
// ---------------------------------------------------------------------------
__global__ void __launch_bounds__(TPB)
coma_main(const float4* __restrict__ gp,
          const float4* __restrict__ gc,
          const float4* __restrict__ gh,
          double* __restrict__ part,
          long long n4) {
  __shared__ float4 sm[2][3][TPB];     // 24 KB: 2 buffers x {prob,c,h} x 256 lanes
  __shared__ float  wpart[TPB / 32];

  const int tid = threadIdx.x;
  long long i   = (long long)blockIdx.x * TPB + tid;

  unsigned lofs[2][3];
#pragma unroll
  for (int b = 0; b < 2; ++b) {
#pragma unroll
    for (int a = 0; a < 3; ++a) lofs[b][a] = lds_byte_off(&sm[b][a][tid]);
  }

  const long long nstages = n4 / STRIDE;   // constant divisor -> shift
  float t = 0.0f;

  if (nstages > 0) {
    // Prologue: stage 0 -> buffer 0 (3 async B128 copies in flight).
    async_copy_b128(gp + i, lofs[0][0]);
    async_copy_b128(gc + i, lofs[0][1]);
    async_copy_b128(gh + i, lofs[0][2]);

    for (long long s = 0; s < nstages; ++s) {
      const long long inext = i + STRIDE;
      if (s + 1 < nstages) {
        const int nb = (int)((s + 1) & 1);
        async_copy_b128(gp + inext, lofs[nb][0]);
        async_copy_b128(gc + inext, lofs[nb][1]);
        async_copy_b128(gh + inext, lofs[nb][2]);
        // 6 outstanding; async loads complete in order -> wait for stage s's 3.
        asm volatile("s_wait_asynccnt 0x3" ::: "memory");
      } else {
        asm volatile("s_wait_asynccnt 0x0" ::: "memory");
      }
      const int    b  = (int)(s & 1);
      const float4 p4 = sm[b][0][tid];
      const float4 c4 = sm[b][1][tid];
      const float4 h4 = sm[b][2][tid];
      t += coma_elem(p4.x, c4.x, h4.x);
      t += coma_elem(p4.y, c4.y, h4.y);
      t += coma_elem(p4.z, c4.z, h4.z);
      t += coma_elem(p4.w, c4.w, h4.w);
      i = inext;
    }
  }

  // Tail (runs only if n4 is not a multiple of the grid stride).
  for (; i < n4; i += STRIDE) {
    const float4 p4 = gp[i];
    const float4 c4 = gc[i];
    const float4 h4 = gh[i];
    t += coma_elem(p4.x, c4.x, h4.x);
    t += coma_elem(p4.y, c4.y, h4.y);
    t += coma_elem(p4.z, c4.z, h4.z);
    t += coma_elem(p4.w, c4.w, h4.w);
  }

  // wave32 tree reduction, then cross-wave combine in LDS.
#pragma unroll
  for (int off = 16; off > 0; off >>= 1) t += __shfl_down(t, off, 32);
  if ((tid & 31) == 0) wpart[tid >> 5] = t;
  __syncthreads();
  if (tid == 0) {
    float s = 0.0f;
#pragma unroll
    for (int w = 0; w < TPB / 32; ++w) s += wpart[w];
    part[blockIdx.x] = (double)s;   // deterministic: fixed slot, no atomics
  }
}

// ---------------------------------------------------------------------------
// Deterministic final reduction of the 2048 block partials + scalar tail.
// ---------------------------------------------------------------------------
__global__ void __launch_bounds__(TPB)
coma_finalize(const double* __restrict__ part, int nparts,
              const float* __restrict__ gp, const float* __restrict__ gc,
              const float* __restrict__ gh, float* __restrict__ out,
              long long n4, long long n) {
  __shared__ double wpart[TPB / 32];
  const int tid = threadIdx.x;

  double a = 0.0;
  for (int i = tid; i < nparts; i += TPB) a += part[i];
#pragma unroll
  for (int off = 16; off > 0; off >>= 1) a += __shfl_down(a, off, 32);
  if ((tid & 31) == 0) wpart[tid >> 5] = a;
  __syncthreads();
  if (tid == 0) {
    double s = 0.0;
#pragma unroll
    for (int w = 0; w < TPB / 32; ++w) s += wpart[w];
    for (long long i = n4 * 4; i < n; ++i)     // <=3 leftover scalars
      s += (double)coma_elem(gp[i], gc[i], gh[i]);
    out[0] = (float)(s * (-1.0 / (double)n));
  }
}

// ---------------------------------------------------------------------------
extern "C" void kernel_launch(void* const* d_in, const int* in_sizes, int n_in,
                              void* d_out, int out_size, void* d_ws, size_t ws_size,
                              hipStream_t stream) {
  (void)n_in; (void)out_size; (void)ws_size;
  const float* p = (const float*)d_in[0];   // prob_map
  const float* c = (const float*)d_in[1];   // c
  const float* h = (const float*)d_in[2];   // h_sampled

  const long long n  = (long long)in_sizes[0];   // B*H*W (all three inputs equal)
  const long long n4 = n >> 2;

  double* part = (double*)d_ws;                  // 2048 * 8B = 16 KB scratch

  coma_main<<<NBLK, TPB, 0, stream>>>((const float4*)p, (const float4*)c,
                                      (const float4*)h, part, n4);
  coma_finalize<<<1, TPB, 0, stream>>>(part, NBLK, p, c, h, (float*)d_out, n4, n);
}